// CBOW_60593398612478
// MI455X (gfx1250) — compile-verified
//
#include <hip/hip_runtime.h>

typedef __attribute__((ext_vector_type(2))) float v2f;
typedef __attribute__((ext_vector_type(8))) float v8f;

namespace {
constexpr int kB = 64;                    // batch rows
constexpr int kL = 2048;                  // sequence length
constexpr int kE = 128;                   // embedding dim
constexpr int kWaves = 16;                // wave32 waves per block
constexpr int kThreads = kWaves * 32;     // 512
constexpr int kTokPerWave = kL / kWaves;  // 128 tokens per wave
}

__global__ __launch_bounds__(kThreads)
void cbow_fused_kernel(const int* __restrict__ x,
                       const float* __restrict__ emb,
                       float* __restrict__ out)
{
    __shared__ int   tokLds[kL];        // 8 KB: this batch row's tokens
    __shared__ float P[kWaves][kE];     // 8 KB: per-wave partial sums (16 x 128)

    const int b    = blockIdx.x;
    const int tid  = threadIdx.x;
    const int wave = tid >> 5;          // wave32
    const int lane = tid & 31;

    // ---- Stage tokens into LDS (streams 8 KB once; later reads are DS broadcasts)
    const int* xrow = x + b * kL;
    #pragma unroll
    for (int i = tid; i < kL; i += kThreads) tokLds[i] = xrow[i];
    __syncthreads();

    // ---- Gather + sum: lane owns embedding dims [4*lane, 4*lane+4).
    // One global_load_b128 per wave pulls a full 512 B embedding row; unroll 8
    // keeps ~4 KB per wave in flight (x1024 waves) to cover L2 gather latency.
    float4 acc = make_float4(0.f, 0.f, 0.f, 0.f);
    const int base = wave * kTokPerWave;
    #pragma unroll 8
    for (int i = 0; i < kTokPerWave; ++i) {
        const int tok = tokLds[base + i];
        const float4 v =
            *reinterpret_cast<const float4*>(emb + (size_t)tok * kE + lane * 4);
        acc.x += v.x; acc.y += v.y; acc.z += v.z; acc.w += v.w;
    }
    P[wave][lane * 4 + 0] = acc.x;
    P[wave][lane * 4 + 1] = acc.y;
    P[wave][lane * 4 + 2] = acc.z;
    P[wave][lane * 4 + 3] = acc.w;
    __syncthreads();

    if (wave == 0) {
        // ---- Cross-wave reduction as WMMA: S[1x128] = ones[1x16] * P[16x128].
        // A = all-ones 16x4 (layout-invariant). B = 4x16 tile of P per ISA
        // striping: lanes 0-15 carry K rows {k0, k0+1} with k0 = g*4 (+2 for
        // lanes 16-31). Every row of D equals S, so d[0] is valid in all lanes.
        const int n     = lane & 15;
        const int khalf = (lane >> 4) << 1;   // 0 for lanes 0-15, 2 for 16-31
        v2f a; a[0] = 1.0f; a[1] = 1.0f;

        const int tFirst0 = tokLds[0];
        const int tFirst1 = tokLds[1];
        const int tLast1  = tokLds[kL - 2];
        const int tLast0  = tokLds[kL - 1];

        float* orow = out + (size_t)b * 4 * kE;

        for (int t = 0; t < 8; ++t) {         // 8 tiles of 16 embedding dims
            v8f d = {};
            #pragma unroll
            for (int g = 0; g < 4; ++g) {     // K = 16 partials, 4 per WMMA
                const int k0 = g * 4 + khalf;
                v2f bb;
                bb[0] = P[k0 + 0][t * 16 + n];
                bb[1] = P[k0 + 1][t * 16 + n];
                d = __builtin_amdgcn_wmma_f32_16x16x4_f32(
                        /*neg_a=*/false, a, /*neg_b=*/false, bb,
                        /*c_mod=*/(short)0, d,
                        /*reuse_a=*/false, /*reuse_b=*/false);
            }
            const float S = d[0];             // row-sum, replicated across rows
            if (lane < 16) {
                const int e = t * 16 + n;
                // Boundary corrections: OOB positions contribute emb[0].
                const float z0 = emb[e];                          // emb[token 0]
                const float zA = emb[(size_t)tFirst0 * kE + e];   // emb[x[b,0]]
                const float zB = emb[(size_t)tFirst1 * kE + e];   // emb[x[b,1]]
                const float zC = emb[(size_t)tLast1  * kE + e];   // emb[x[b,L-2]]
                const float zD = emb[(size_t)tLast0  * kE + e];   // emb[x[b,L-1]]
                orow[0 * kE + e] = S - zD + z0;                   // offset -1
                orow[1 * kE + e] = S - zD - zC + 2.f * z0;        // offset -2
                orow[2 * kE + e] = S - zA + z0;                   // offset +1
                orow[3 * kE + e] = S - zA - zB + 2.f * z0;        // offset +2
            }
        }
    }
}

extern "C" void kernel_launch(void* const* d_in, const int* in_sizes, int n_in,
                              void* d_out, int out_size, void* d_ws, size_t ws_size,
                              hipStream_t stream) {
    (void)in_sizes; (void)n_in; (void)d_ws; (void)ws_size; (void)out_size;
    const int*   x   = (const int*)d_in[0];    // token ids (int32)
    const float* emb = (const float*)d_in[1];  // [100000, 128] f32
    float*       out = (float*)d_out;          // [64, 4, 128] f32
    cbow_fused_kernel<<<dim3(kB), dim3(kThreads), 0, stream>>>(x, emb, out);
}